// CrowdCountingLoss_17961553231887
// MI455X (gfx1250) — compile-verified
//
#include <hip/hip_runtime.h>
#include <hip/hip_bf16.h>
#include <math.h>

// CrowdCountingLoss on gfx1250.
// Sinkhorn with separable Gaussian kernel: each softmin = 2x (64x64x64) f32
// matmuls done with V_WMMA_F32_16X16X4_F32, fully LDS-resident, single WGP.

#define NPIX 4096     // 64*64
#define GDIM 64
#define NTHREADS 256  // 8 wave32s

typedef float v2f __attribute__((ext_vector_type(2)));
typedef float v8f __attribute__((ext_vector_type(8)));

__device__ __forceinline__ float waveRedMax(float v) {
#pragma unroll
  for (int o = 16; o > 0; o >>= 1) v = fmaxf(v, __shfl_xor(v, o, 32));
  return v;
}
__device__ __forceinline__ float waveRedSum(float v) {
#pragma unroll
  for (int o = 16; o > 0; o >>= 1) v += __shfl_xor(v, o, 32);
  return v;
}

__device__ __forceinline__ float blockMax(float v, float* red) {
  v = waveRedMax(v);
  if ((threadIdx.x & 31) == 0) red[threadIdx.x >> 5] = v;
  __syncthreads();
  if (threadIdx.x == 0) {
    float m = red[0];
#pragma unroll
    for (int i = 1; i < 8; ++i) m = fmaxf(m, red[i]);
    red[8] = m;
  }
  __syncthreads();
  float r = red[8];
  __syncthreads();
  return r;
}

__device__ __forceinline__ float blockSum(float v, float* red) {
  v = waveRedSum(v);
  if ((threadIdx.x & 31) == 0) red[threadIdx.x >> 5] = v;
  __syncthreads();
  if (threadIdx.x == 0) {
    float s = red[0];
#pragma unroll
    for (int i = 1; i < 8; ++i) s += red[i];
    red[8] = s;
  }
  __syncthreads();
  float r = red[8];
  __syncthreads();
  return r;
}

// C = A @ B, all 64x64 row-major f32 in LDS. 8 waves, 2 output tiles each.
// Fragment layouts per CDNA5 ISA 7.12.2:
//   A (16x4, 2 VGPR): lane half h, lane l=lane&15 -> M = l, K = k0 + 2h + r
//   B (4x16, 2 VGPR): K = k0 + 2h + r, N = l
//   C/D (16x16, 8 VGPR): reg r, half h -> M = r + 8h, N = l
__device__ __forceinline__ void mm64_wmma(const float* __restrict__ A,
                                          const float* __restrict__ B,
                                          float* __restrict__ C) {
  const int lane = threadIdx.x & 31;
  const int wave = threadIdx.x >> 5;
  const int h = lane >> 4;
  const int l = lane & 15;
#pragma unroll
  for (int rep = 0; rep < 2; ++rep) {
    const int t = wave * 2 + rep;       // 16 tiles total (4x4 grid of 16x16)
    const int ti = t >> 2, tj = t & 3;
    v8f acc = {};
#pragma unroll
    for (int k0 = 0; k0 < GDIM; k0 += 4) {
      v2f af, bf;
      const float* ap = A + (ti * 16 + l) * GDIM + k0 + 2 * h;
      af.x = ap[0];
      af.y = ap[1];
      const float* bp = B + (k0 + 2 * h) * GDIM + tj * 16 + l;
      bf.x = bp[0];
      bf.y = bp[GDIM];
      acc = __builtin_amdgcn_wmma_f32_16x16x4_f32(
          /*neg_a=*/false, af, /*neg_b=*/false, bf,
          /*c_mod=*/(short)0, acc, /*reuse_a=*/false, /*reuse_b=*/false);
    }
#pragma unroll
    for (int r = 0; r < 8; ++r)
      C[(ti * 16 + r + 8 * h) * GDIM + tj * 16 + l] = acc[r];
  }
}

__device__ __forceinline__ void buildG(float eps, float* G) {
  const float inv2e = 0.5f / eps;
  for (int i = threadIdx.x; i < NPIX; i += NTHREADS) {
    const int r = i >> 6, c = i & 63;
    const float d = (float)(r - c);
    G[i] = expf(-d * d * inv2e);
  }
  __syncthreads();
}

// out[i] = lam * softmin(eps, C, baseLog + hpot/eps)[i], optionally averaged
// with avgWith[i] (the geomloss symmetric update). Uses separable kernel:
// S = G @ exp(h - m) @ G  (G symmetric), softmin_i = -eps*(m + log S_i).
__device__ void softmin_sep(float eps, float lam, const float* baseLog,
                            const float* hpot, const float* avgWith, float* out,
                            const float* G, float* Wk, float* Tm, float* red) {
  const int tid = threadIdx.x;
  const float invE = 1.0f / eps;
  float lmax = -3.0e38f;
  for (int i = tid; i < NPIX; i += NTHREADS) {
    float hv = baseLog[i];
    if (hpot) hv += hpot[i] * invE;
    Wk[i] = hv;
    lmax = fmaxf(lmax, hv);
  }
  __syncthreads();
  const float m = blockMax(lmax, red);
  for (int i = tid; i < NPIX; i += NTHREADS) Wk[i] = expf(Wk[i] - m);
  __syncthreads();
  mm64_wmma(Wk, G, Tm);  // Tm = E @ G (== E @ G^T, G symmetric)
  __syncthreads();
  mm64_wmma(G, Tm, Wk);  // Wk = G @ Tm = S
  __syncthreads();
  for (int i = tid; i < NPIX; i += NTHREADS) {
    const float s = lam * (-eps) * (m + logf(fmaxf(Wk[i], 1.0e-38f)));
    out[i] = avgWith ? 0.5f * (avgWith[i] + s) : s;
  }
  __syncthreads();
}

__global__ __launch_bounds__(NTHREADS) void crowd_counting_loss_kernel(
    const float* __restrict__ pred, const float* __restrict__ gt,
    float* __restrict__ out) {
  extern __shared__ float sm[];
  float* a   = sm;            // raw pred
  float* b   = a + NPIX;      // raw gt
  float* aL  = b + NPIX;      // log a
  float* bL  = aL + NPIX;     // log b
  float* fba = bL + NPIX;     // potentials
  float* gab = fba + NPIX;
  float* faa = gab + NPIX;
  float* gbb = faa + NPIX;
  float* G   = gbb + NPIX;    // separable Gaussian kernel for current eps
  float* Wk  = G + NPIX;      // h / E / S workspace
  float* Tm  = Wk + NPIX;     // matmul intermediate
  float* tmp = Tm + NPIX;     // stash for f_ba update (cross-coupling)
  float* red = tmp + NPIX;    // 16-float reduction scratch

  const int tid = threadIdx.x;
  for (int i = tid; i < NPIX; i += NTHREADS) {
    const float av = pred[i], bv = gt[i];
    a[i] = av;
    b[i] = bv;
    aL[i] = logf(av);
    bL[i] = logf(bv);
  }
  __syncthreads();

  const float blur = 0.2f, scaling = 0.9f, reach = 0.1f, alpha = 0.1f;
  const float rho = reach * reach;                    // 0.01
  const float diam2 = 63.0f * 63.0f * 2.0f;           // diameter^p
  const float startL = logf(diam2);                   // p*log(diam)
  const float stopL = 2.0f * logf(blur);
  const float stepL = 2.0f * logf(scaling);
  int nA = 0;
  for (float t = startL; t > stopL; t += stepL) nA++;
  const int L = nA + 2;  // [diam^2] + arange + [blur^2]

  // ---- init potentials at eps0 = diam^2 ----
  {
    const float e0 = diam2, lam0 = rho / (rho + e0);
    buildG(e0, G);
    softmin_sep(e0, lam0, bL, nullptr, nullptr, fba, G, Wk, Tm, red);
    softmin_sep(e0, lam0, aL, nullptr, nullptr, gab, G, Wk, Tm, red);
    for (int i = tid; i < NPIX; i += NTHREADS) {
      faa[i] = gab[i];  // softmin(e0, a_log)
      gbb[i] = fba[i];  // softmin(e0, b_log)
    }
    __syncthreads();
  }

  // ---- annealed symmetric Sinkhorn scan ----
  for (int it = 0; it < L; ++it) {
    const float eps = (it == 0) ? diam2
                     : ((it == L - 1) ? blur * blur
                                      : expf(startL + (float)(it - 1) * stepL));
    const float lam = rho / (rho + eps);
    buildG(eps, G);
    // f_ba_n = lam*softmin(eps, b_log + g_ab/eps); avg with old f_ba -> tmp
    softmin_sep(eps, lam, bL, gab, fba, tmp, G, Wk, Tm, red);
    // g_ab_n = lam*softmin(eps, a_log + f_ba/eps); avg with old g_ab (in place)
    softmin_sep(eps, lam, aL, fba, gab, gab, G, Wk, Tm, red);
    // self-coupled debias terms, in place
    softmin_sep(eps, lam, aL, faa, faa, faa, G, Wk, Tm, red);
    softmin_sep(eps, lam, bL, gbb, gbb, gbb, G, Wk, Tm, red);
    for (int i = tid; i < NPIX; i += NTHREADS) fba[i] = tmp[i];
    __syncthreads();
  }

  // ---- final extrapolation at eps = blur^2, then the debiased cost ----
  {
    const float eps = blur * blur;
    const float lam = rho / (rho + eps);
    buildG(eps, G);
    softmin_sep(eps, lam, bL, gab, nullptr, tmp, G, Wk, Tm, red);  // f_ba_f
    softmin_sep(eps, lam, aL, fba, nullptr, gab, G, Wk, Tm, red);  // g_ab_f
    softmin_sep(eps, lam, aL, faa, nullptr, faa, G, Wk, Tm, red);  // f_aa_f
    softmin_sep(eps, lam, bL, gbb, nullptr, gbb, G, Wk, Tm, red);  // g_bb_f

    const float invR = 1.0f / rho;
    float part = 0.0f, d2 = 0.0f, sa = 0.0f, sb = 0.0f;
    for (int i = tid; i < NPIX; i += NTHREADS) {
      part += a[i] * (expf(-faa[i] * invR) - expf(-tmp[i] * invR)) +
              b[i] * (expf(-gbb[i] * invR) - expf(-gab[i] * invR));
      const float d = a[i] - b[i];
      d2 += d * d;
      sa += a[i];
      sb += b[i];
    }
    const float spatial = blockSum(part, red);
    const float mse = blockSum(d2, red) * (1.0f / (float)NPIX);
    const float cnt = fabsf(blockSum(sa, red) - blockSum(sb, red));
    if (tid == 0) {
      const float w = rho + 0.5f * eps;
      out[0] = mse + cnt + alpha * (w * spatial);
    }
  }
}

extern "C" void kernel_launch(void* const* d_in, const int* in_sizes, int n_in,
                              void* d_out, int out_size, void* d_ws,
                              size_t ws_size, hipStream_t stream) {
  (void)in_sizes; (void)n_in; (void)d_ws; (void)ws_size; (void)out_size;
  const float* pred = (const float*)d_in[0];     // [64,64]
  const float* gt = (const float*)d_in[1];       // [1,1,64,64]
  // d_in[2] (gt_blur_map) does not affect the reference loss.
  float* out = (float*)d_out;

  // 12 * 16KB matrices + reduction scratch, all resident in one WGP's LDS.
  const size_t shmem = (size_t)(12 * NPIX + 16) * sizeof(float);
  crowd_counting_loss_kernel<<<1, NTHREADS, shmem, stream>>>(pred, gt, out);
}